// NSLoss_6923487281350
// MI455X (gfx1250) — compile-verified
//
#include <hip/hip_runtime.h>

typedef __attribute__((ext_vector_type(2))) float v2f;
typedef __attribute__((ext_vector_type(8))) float v8f;

// Numerically stable log_sigmoid(x) = min(x,0) - log(1 + exp(-|x|))
__device__ __forceinline__ float log_sigmoid(float x) {
    float ax = __builtin_fabsf(x);
    return fminf(x, 0.0f) - __logf(1.0f + __expf(-ax));
}

// Dynamic-index extraction from an 8-wide accumulator (lowers to v_cndmask chain)
__device__ __forceinline__ float sel8(v8f c, int r) {
    float x = c[0];
#pragma unroll
    for (int k = 1; k < 8; ++k) x = (r == k) ? c[k] : x;
    return x;
}

__global__ void nsloss_zero(float* acc) {
    if (threadIdx.x == 0) acc[0] = 0.0f;
}

__global__ __launch_bounds__(256) void nsloss_main(
    const float* __restrict__ embs,     // [N,128] fp32
    const int*   __restrict__ label,    // [N]
    const int*   __restrict__ negs,     // [N,5]
    const float* __restrict__ weights,  // [1M,128] fp32
    float* __restrict__ acc, int ntiles)
{
    const int lane = threadIdx.x & 31;
    const int wave = threadIdx.x >> 5;
    const int tile = blockIdx.x * (blockDim.x >> 5) + wave;
    if (tile >= ntiles) return;          // wave-uniform: EXEC stays all-1s for WMMA

    const int col  = lane & 15;          // matrix row (A) / column (B) owned by this lane
    const int half = lane >> 4;          // K-half select: lanes 16-31 carry K = {2,3} mod 4
    const int row  = tile * 16 + col;

    // A operand: embs rows. Per K-step kk, lane loads embs[row][4kk + 2*half + {0,1}].
    const float* eptr = embs + (long)row * 128 + 2 * half;
    // B operands: gathered weight rows (column `col` of each B), same K addressing.
    const float* w0 = weights + (long)label[row]        * 128 + 2 * half; // positive
    const float* w1 = weights + (long)negs[row * 5 + 0] * 128 + 2 * half;
    const float* w2 = weights + (long)negs[row * 5 + 1] * 128 + 2 * half;
    const float* w3 = weights + (long)negs[row * 5 + 2] * 128 + 2 * half;
    const float* w4 = weights + (long)negs[row * 5 + 3] * 128 + 2 * half;
    const float* w5 = weights + (long)negs[row * 5 + 4] * 128 + 2 * half;

    v8f c0 = {}, c1 = {}, c2 = {}, c3 = {}, c4 = {}, c5 = {};

#pragma unroll 4
    for (int kk = 0; kk < 32; ++kk) {
        const int o = kk * 4;
        v2f a  = *(const v2f*)(eptr + o);
        v2f b0 = *(const v2f*)(w0 + o);
        v2f b1 = *(const v2f*)(w1 + o);
        v2f b2 = *(const v2f*)(w2 + o);
        v2f b3 = *(const v2f*)(w3 + o);
        v2f b4 = *(const v2f*)(w4 + o);
        v2f b5 = *(const v2f*)(w5 + o);
        // D = A x B + C  (16x16x4 fp32); 8 args: (neg_a, A, neg_b, B, c_mod, C, reuse_a, reuse_b)
        c0 = __builtin_amdgcn_wmma_f32_16x16x4_f32(false, a, false, b0, (short)0, c0, false, false);
        c1 = __builtin_amdgcn_wmma_f32_16x16x4_f32(false, a, false, b1, (short)0, c1, false, false);
        c2 = __builtin_amdgcn_wmma_f32_16x16x4_f32(false, a, false, b2, (short)0, c2, false, false);
        c3 = __builtin_amdgcn_wmma_f32_16x16x4_f32(false, a, false, b3, (short)0, c3, false, false);
        c4 = __builtin_amdgcn_wmma_f32_16x16x4_f32(false, a, false, b4, (short)0, c4, false, false);
        c5 = __builtin_amdgcn_wmma_f32_16x16x4_f32(false, a, false, b5, (short)0, c5, false, false);
    }

    // Diagonal of D: d<8 at (VGPR d, lane d); d>=8 at (VGPR d-8, lane d+16).
    const int  reg = (lane < 8) ? lane : (lane - 24);
    const float m  = ((lane < 8) || (lane >= 24)) ? 1.0f : 0.0f;

    float L = log_sigmoid( sel8(c0, reg));   // positive term
    L +=      log_sigmoid(-sel8(c1, reg));   // noise = -w  ->  logsig(-dot)
    L +=      log_sigmoid(-sel8(c2, reg));
    L +=      log_sigmoid(-sel8(c3, reg));
    L +=      log_sigmoid(-sel8(c4, reg));
    L +=      log_sigmoid(-sel8(c5, reg));
    L *= m;

#pragma unroll
    for (int off = 16; off > 0; off >>= 1) L += __shfl_xor(L, off, 32);
    if (lane == 0) atomicAdd(acc, L);
}

__global__ void nsloss_final(const float* acc, float* out, float inv_n) {
    if (threadIdx.x == 0) out[0] = -acc[0] * inv_n;
}

extern "C" void kernel_launch(void* const* d_in, const int* in_sizes, int n_in,
                              void* d_out, int out_size, void* d_ws, size_t ws_size,
                              hipStream_t stream) {
    const float* embs    = (const float*)d_in[1];
    const int*   label   = (const int*)  d_in[2];
    const int*   negs    = (const int*)  d_in[3];
    const float* weights = (const float*)d_in[4];
    const int n = in_sizes[0];          // 65536 (multiple of 16)

    float* acc = (float*)d_ws;
    nsloss_zero<<<1, 1, 0, stream>>>(acc);

    const int ntiles = n / 16;          // one wave32 per 16-row tile
    const int waves_per_block = 8;      // 256 threads = 8 waves
    const int blocks = (ntiles + waves_per_block - 1) / waves_per_block;
    nsloss_main<<<blocks, 256, 0, stream>>>(embs, label, negs, weights, acc, ntiles);

    nsloss_final<<<1, 1, 0, stream>>>(acc, (float*)d_out, 1.0f / (float)n);
}